// EdgeSpecificTreeLSTMCell_39273180954989
// MI455X (gfx1250) — compile-verified
//
#include <hip/hip_runtime.h>
#include <math.h>

// ---------------- problem constants (from reference) ----------------
#define NLEV   6
#define NTOT   52500
#define NEDGE  52000
#define XS     128          // X_SIZE
#define HS     64           // H_SIZE
#define IOUW   192          // 3*H
#define ACCW   256          // 192 (iou) + 64 (x_f)

static const int NOFF_h[NLEV + 1] = {0, 500, 2500, 8500, 20500, 36500, 52500};

typedef __attribute__((ext_vector_type(2))) float v2f;
typedef __attribute__((ext_vector_type(8))) float v8f;

__device__ __forceinline__ float sigmoidf(float v) {
    return 1.0f / (1.0f + __expf(-v));
}

__device__ __forceinline__ int level_of_edge(int e) {
    // level-l edges occupy [NOFF[l]-500, NOFF[l+1]-500)
    return (e >= 36000) ? 5 : (e >= 20000) ? 4 : (e >= 8000) ? 3 : (e >= 2000) ? 2 : 1;
}

// ---------------- workspace zeroing ----------------
__global__ void zero_kernel(float* __restrict__ p, size_t n) {
    size_t i = (size_t)blockIdx.x * blockDim.x + threadIdx.x;
    size_t stride = (size_t)gridDim.x * blockDim.x;
    for (; i < n; i += stride) p[i] = 0.0f;
}

// ---------------- binning: count edges per (level, matrix_id) ----------------
__global__ void count_kernel(const int* __restrict__ matrix_id, int* __restrict__ cnt) {
    int e = blockIdx.x * blockDim.x + threadIdx.x;
    if (e >= NEDGE) return;
    int li = level_of_edge(e) - 1;               // 0..4
    atomicAdd(&cnt[li * 32 + matrix_id[e]], 1);
}

// single-thread prefix sums: offs[li*33+m], toffs[li*33+m] (tiles of 16), cursor init
__global__ void prefix_kernel(const int* __restrict__ cnt,
                              int* __restrict__ offs, int* __restrict__ toffs,
                              int* __restrict__ cursor) {
    if (threadIdx.x != 0 || blockIdx.x != 0) return;
    for (int li = 0; li < 5; ++li) {
        int s = 0, ts = 0;
        for (int m = 0; m < 32; ++m) {
            offs [li * 33 + m] = s;
            toffs[li * 33 + m] = ts;
            cursor[li * 32 + m] = s;
            int c = cnt[li * 32 + m];
            s  += c;
            ts += (c + 15) >> 4;
        }
        offs [li * 33 + 32] = s;
        toffs[li * 33 + 32] = ts;
    }
}

__global__ void scatter_kernel(const int* __restrict__ matrix_id,
                               int* __restrict__ cursor, int* __restrict__ perm) {
    int e = blockIdx.x * blockDim.x + threadIdx.x;
    if (e >= NEDGE) return;
    int lvl = level_of_edge(e);
    int li  = lvl - 1;
    static const int EOFF[5] = {0, 2000, 8000, 20000, 36000};
    int pos = atomicAdd(&cursor[li * 32 + matrix_id[e]], 1);
    perm[EOFF[li] + pos] = e;                     // pos already includes group offset
}

// ---------------- WMMA GEMM: acc[m][0:256] = x[m] @ [W_iou; W_f]^T + bias ----
__global__ void gemm_xw_kernel(const float* __restrict__ x,
                               const float* __restrict__ Wiou,
                               const float* __restrict__ biou,
                               const float* __restrict__ Wf,
                               const float* __restrict__ bf,
                               float* __restrict__ acc,
                               int mtiles) {
    int wave = (int)((blockIdx.x * blockDim.x + threadIdx.x) >> 5);
    int lane = threadIdx.x & 31;
    int ntile = wave & 15;
    int mtile = wave >> 4;
    if (mtile >= mtiles) return;

    int l16 = lane & 15;
    int hlf = lane >> 4;
    int m0  = mtile * 16;
    int arow = m0 + l16;
    if (arow >= NTOT) arow = NTOT - 1;
    int n = ntile * 16 + l16;

    const float* wrow = (n < IOUW) ? (Wiou + (size_t)n * XS)
                                   : (Wf + (size_t)(n - IOUW) * XS);
    const float* ap = x + (size_t)arow * XS + 2 * hlf;
    const float* bp = wrow + 2 * hlf;

    v8f c = {};
    #pragma unroll 4
    for (int k0 = 0; k0 < XS; k0 += 4) {
        v2f a = *(const v2f*)(ap + k0);
        v2f b = *(const v2f*)(bp + k0);
        c = __builtin_amdgcn_wmma_f32_16x16x4_f32(false, a, false, b,
                                                  (short)0, c, false, false);
    }

    float bias = (n < IOUW) ? biou[n] : bf[n - IOUW];
    if (m0 + 16 <= NTOT) {
        #pragma unroll
        for (int v = 0; v < 8; ++v)
            acc[(size_t)(m0 + v + 8 * hlf) * ACCW + n] = c[v] + bias;
    } else {
        #pragma unroll
        for (int v = 0; v < 8; ++v) {
            int m = m0 + v + 8 * hlf;
            if (m < NTOT) acc[(size_t)m * ACCW + n] = c[v] + bias;
        }
    }
}

// ---------------- per-level node kernel: h,c from acc/fc_sum ----------------
__global__ void node_kernel(const float* __restrict__ acc,
                            const float* __restrict__ fc_sum,
                            float* __restrict__ out_h,
                            float* __restrict__ out_c,
                            int s0, int s1) {
    int wave = (int)((blockIdx.x * blockDim.x + threadIdx.x) >> 5);
    int lane = threadIdx.x & 31;
    int node = s0 + wave;
    if (node >= s1) return;

    const float* a = acc + (size_t)node * ACCW;
    float i0 = a[lane],       i1 = a[lane + 32];
    float o0 = a[64 + lane],  o1 = a[96 + lane];
    float u0 = a[128 + lane], u1 = a[160 + lane];
    float fs0 = fc_sum[(size_t)node * HS + lane];
    float fs1 = fc_sum[(size_t)node * HS + lane + 32];

    float c0 = sigmoidf(i0) * tanhf(u0) + fs0;
    float c1 = sigmoidf(i1) * tanhf(u1) + fs1;
    float h0 = sigmoidf(o0) * tanhf(c0);
    float h1 = sigmoidf(o1) * tanhf(c1);

    out_h[(size_t)node * HS + lane]      = h0;
    out_h[(size_t)node * HS + lane + 32] = h1;
    out_c[(size_t)node * HS + lane]      = c0;
    out_c[(size_t)node * HS + lane + 32] = c1;
}

// ---------------- per-level edge WMMA kernel ----------------
// One wave per 16-edge tile (edges share matrix_id). A = h[src] rows (16x64 f32)
// held in 16 v2f registers; 12 Uh tiles (16x192) + 4 hf tiles (16x64), each tile
// = 16 chained fp32 WMMAs (K stepped by 4). Stores masked for partial tiles.
__global__ void edge_wmma_kernel(const float* __restrict__ out_h,
                                 const float* __restrict__ out_c,
                                 float* acc, float* fc_sum,
                                 const int* __restrict__ edge_dst,
                                 const float* __restrict__ Uiou,  // [32][64][192]
                                 const float* __restrict__ Uf,    // [32][64][64]
                                 const int* __restrict__ offs,
                                 const int* __restrict__ toffs,
                                 const int* __restrict__ perm,
                                 int li, int eb) {
    int wave = (int)((blockIdx.x * blockDim.x + threadIdx.x) >> 5);
    int lane = threadIdx.x & 31;
    const int* offsL  = offs  + li * 33;
    const int* toffsL = toffs + li * 33;
    int totalTiles = toffsL[32];
    if (wave >= totalTiles) return;

    // find group (matrix id): largest m with toffsL[m] <= wave
    int mid = 0;
    for (int m = 1; m < 32; ++m)
        if (toffsL[m] <= wave) mid = m;
    int ti    = wave - toffsL[mid];
    int base  = offsL[mid];
    int cntm  = offsL[mid + 1] - base;
    int row0  = ti * 16;
    int valid = cntm - row0; if (valid > 16) valid = 16;

    int l16 = lane & 15;
    int hlf = lane >> 4;

    // gather A rows: h[src] for this lane's edge row (clamped for partial tiles)
    int r   = (l16 < valid) ? l16 : (valid - 1);
    int e   = perm[eb + base + row0 + r];
    int src = 500 + e;
    const float* hp = out_h + (size_t)src * HS + 2 * hlf;
    v2f a[16];
    #pragma unroll
    for (int kk = 0; kk < 16; ++kk) a[kk] = *(const v2f*)(hp + kk * 4);

    const float* UiM = Uiou + (size_t)mid * HS * IOUW;
    const float* UfM = Uf   + (size_t)mid * HS * HS;

    // ---- 12 tiles of Uh = hs @ U_iou[mid]  (16 x 192) ----
    for (int j = 0; j < 12; ++j) {
        int n = j * 16 + l16;
        const float* bp = UiM + (size_t)(2 * hlf) * IOUW + n;
        v8f d = {};
        #pragma unroll
        for (int kk = 0; kk < 16; ++kk) {
            v2f b;
            b.x = bp[(size_t)(4 * kk) * IOUW];
            b.y = bp[(size_t)(4 * kk + 1) * IOUW];
            d = __builtin_amdgcn_wmma_f32_16x16x4_f32(false, a[kk], false, b,
                                                      (short)0, d, false, false);
        }
        #pragma unroll
        for (int v = 0; v < 8; ++v) {
            int rr = v + 8 * hlf;
            if (rr < valid) {
                int ee  = perm[eb + base + row0 + rr];
                int dst = edge_dst[ee];
                atomicAdd(acc + (size_t)dst * ACCW + n, d[v]);
            }
        }
    }

    // ---- 4 tiles of hf = hs @ U_f[mid]  (16 x 64), then f/fc scatter ----
    for (int j = 0; j < 4; ++j) {
        int n = j * 16 + l16;
        const float* bp = UfM + (size_t)(2 * hlf) * HS + n;
        v8f d = {};
        #pragma unroll
        for (int kk = 0; kk < 16; ++kk) {
            v2f b;
            b.x = bp[(size_t)(4 * kk) * HS];
            b.y = bp[(size_t)(4 * kk + 1) * HS];
            d = __builtin_amdgcn_wmma_f32_16x16x4_f32(false, a[kk], false, b,
                                                      (short)0, d, false, false);
        }
        #pragma unroll
        for (int v = 0; v < 8; ++v) {
            int rr = v + 8 * hlf;
            if (rr < valid) {
                int ee   = perm[eb + base + row0 + rr];
                int dst  = edge_dst[ee];
                int srcr = 500 + ee;
                float f  = sigmoidf(acc[(size_t)dst * ACCW + IOUW + n] + d[v]);
                float fc = out_c[(size_t)srcr * HS + n] * f;
                atomicAdd(fc_sum + (size_t)dst * HS + n, fc);
            }
        }
    }
}

// ---------------- host-side launcher ----------------
extern "C" void kernel_launch(void* const* d_in, const int* in_sizes, int n_in,
                              void* d_out, int out_size, void* d_ws, size_t ws_size,
                              hipStream_t stream) {
    const float* x         = (const float*)d_in[0];
    const int*   edge_dst  = (const int*)d_in[2];
    const int*   matrix_id = (const int*)d_in[3];
    const float* Wiou      = (const float*)d_in[4];
    const float* biou      = (const float*)d_in[5];
    const float* Wf        = (const float*)d_in[6];
    const float* bf        = (const float*)d_in[7];
    const float* Uiou      = (const float*)d_in[8];
    const float* Uf        = (const float*)d_in[9];

    float* out_h = (float*)d_out;
    float* out_c = out_h + (size_t)NTOT * HS;

    // workspace layout
    float* acc    = (float*)d_ws;                      // [NTOT][256]
    float* fc_sum = acc + (size_t)NTOT * ACCW;         // [NTOT][64]
    int*   cnt    = (int*)(fc_sum + (size_t)NTOT * HS);// [5][32]
    int*   offs   = cnt + 5 * 32;                      // [5][33]
    int*   toffs  = offs + 5 * 33;                     // [5][33]
    int*   cursor = toffs + 5 * 33;                    // [5][32]
    int*   perm   = cursor + 5 * 32;                   // [52000]

    // zero fc_sum and cnt (contiguous region)
    zero_kernel<<<512, 256, 0, stream>>>(fc_sum, (size_t)NTOT * HS + 5 * 32);

    // binning: counts -> prefix -> scatter
    count_kernel <<<(NEDGE + 255) / 256, 256, 0, stream>>>(matrix_id, cnt);
    prefix_kernel<<<1, 32, 0, stream>>>(cnt, offs, toffs, cursor);
    scatter_kernel<<<(NEDGE + 255) / 256, 256, 0, stream>>>(matrix_id, cursor, perm);

    // x projections via WMMA GEMM
    int mtiles = (NTOT + 15) / 16;
    int waves  = mtiles * 16;
    gemm_xw_kernel<<<(waves + 7) / 8, 256, 0, stream>>>(x, Wiou, biou, Wf, bf, acc, mtiles);

    // leaf -> root
    for (int l = NLEV - 1; l >= 0; --l) {
        int s0 = NOFF_h[l], s1 = NOFF_h[l + 1];
        int nn = s1 - s0;
        node_kernel<<<(nn + 7) / 8, 256, 0, stream>>>(acc, fc_sum, out_h, out_c, s0, s1);
        if (l > 0) {
            int eb = NOFF_h[l] - 500;                  // level's base in perm/edge arrays
            int maxTiles = nn / 16 + 32;               // upper bound: sum ceil(cnt/16)
            edge_wmma_kernel<<<(maxTiles + 7) / 8, 256, 0, stream>>>(
                out_h, out_c, acc, fc_sum, edge_dst, Uiou, Uf,
                offs, toffs, perm, l - 1, eb);
        }
    }
}